// PhoenixEmulator_68728066671257
// MI455X (gfx1250) — compile-verified
//
#include <hip/hip_runtime.h>
#include <math.h>

// ---------------------------------------------------------------------------
// PhoenixEmulator: sum of 1024 Voigt profiles over 20000 wavelengths,
// times blackbody * quadratic modulation.  Compute-bound transcendental
// workload; optimized via per-line precompute + sorted-center window pruning
// (|x|>25 underflows to exact f32 zero) + CDNA5 async global->LDS staging.
// ---------------------------------------------------------------------------

#define PI_F      3.14159265358979f
#define SQRT2_F   1.41421356237309f
#define SQRT2PI_F 2.50662827463100f
#define XCUT      25.0f
#define XCUT2     (XCUT * XCUT)
#define WS_HDR    64   // floats reserved at head of workspace (maxhw at [0])

__constant__ float S1_C[8] = {0.778800786f, 0.36787945f, 0.105399221f,
                              0.0183156393f, 0.00193045416f, 0.000123409802f,
                              4.78511765e-06f, 1.12535176e-07f};
__constant__ float S1_K[8] = {0.25f, 1.0f, 2.25f, 4.0f, 6.25f, 9.0f, 12.25f, 16.0f};

// ---------------------------------------------------------------------------
// Kernel 1: per-line precompute (single block).
// blob layout (SoA, each length L): lam, inv=1/(sqrt2*sig), y, y2, A, B, C
//   voigt(l,w) = exx*(cos2xy*A + (1-cos2xy)*B) + C*sigma23
//   A = pref*(erfcx(y) - y*s1sum/pi),  B = pref/(2*pi*y),  C = pref*y/(2*pi)
// Also block-reduces max window half-width (XCUT/inv) into ws[0].
// ---------------------------------------------------------------------------
__global__ void phoenix_prep(const float* __restrict__ lam,
                             const float* __restrict__ log_amp,
                             const float* __restrict__ log_sig,
                             const float* __restrict__ log_gam,
                             float* __restrict__ ws, int L) {
  __shared__ float red[256];
  float* blob = ws + WS_HDR;
  float hwmax = 0.0f;
  for (int l = threadIdx.x; l < L; l += blockDim.x) {
    float sig  = expf(log_sig[l]);
    float gam  = expf(log_gam[l]);
    float amp  = expf(log_amp[l]);
    float inv  = 1.0f / (SQRT2_F * sig);
    float y    = gam * inv;
    float y2   = y * y;
    float pref = amp / (SQRT2PI_F * sig);
    float erfcxy = expf(y2) * erfcf(y);
    float s1 = 0.0f;
#pragma unroll
    for (int k = 0; k < 8; ++k) s1 += S1_C[k] / (S1_K[k] + y2);
    float A = pref * (erfcxy - y * s1 / PI_F);
    float B = pref / (2.0f * PI_F * y);
    float C = pref * y / (2.0f * PI_F);
    blob[0 * L + l] = lam[l];
    blob[1 * L + l] = inv;
    blob[2 * L + l] = y;
    blob[3 * L + l] = y2;
    blob[4 * L + l] = A;
    blob[5 * L + l] = B;
    blob[6 * L + l] = C;
    hwmax = fmaxf(hwmax, XCUT / inv);
  }
  red[threadIdx.x] = hwmax;
  __syncthreads();
  for (int s = blockDim.x >> 1; s > 0; s >>= 1) {
    if ((int)threadIdx.x < s) red[threadIdx.x] = fmaxf(red[threadIdx.x], red[threadIdx.x + s]);
    __syncthreads();
  }
  if (threadIdx.x == 0) ws[0] = red[0];
}

// ---------------------------------------------------------------------------
// Kernel 2: one thread per wavelength.  Stages the 7*L-float parameter blob
// into LDS with gfx1250 async global->LDS b128 loads (ASYNCcnt tracked),
// binary-searches the sorted line centers for the active window, accumulates
// the Voigt sum, then applies blackbody * modulation.
// ---------------------------------------------------------------------------
__global__ void phoenix_main(const float* __restrict__ wl,
                             const float* __restrict__ ws,
                             const float* __restrict__ ln_teff,
                             const float* __restrict__ a_c,
                             const float* __restrict__ b_c,
                             const float* __restrict__ c_c,
                             float* __restrict__ out, int W, int L) {
  extern __shared__ float smem[];           // 7*L floats (+pad)
  const float* blob = ws + WS_HDR;
  const int tid = threadIdx.x;
  const int nfloats  = 7 * L;
  const int nchunks4 = nfloats >> 2;        // full 16-byte chunks

  // --- async stage blob -> LDS (CDNA5 path: global_load_async_to_lds_b128) ---
  for (int c = tid; c < nchunks4; c += blockDim.x) {
    unsigned lds_off = (unsigned)(uintptr_t)(&smem[c * 4]);
    const float* g = blob + c * 4;
    asm volatile("global_load_async_to_lds_b128 %0, %1, off"
                 :: "v"(lds_off), "v"(g)
                 : "memory");
  }
  int tail = nfloats & 3;                   // dead for L multiple of 4
  if (tid < tail) smem[nchunks4 * 4 + tid] = blob[nchunks4 * 4 + tid];
  asm volatile("s_wait_asynccnt 0" ::: "memory");
  __syncthreads();

  const float* lamS = smem;
  const float* invS = smem + 1 * L;
  const float* yS   = smem + 2 * L;
  const float* y2S  = smem + 3 * L;
  const float* AS   = smem + 4 * L;
  const float* BS   = smem + 5 * L;
  const float* CS   = smem + 6 * L;

  int w = blockIdx.x * blockDim.x + tid;
  if (w >= W) return;

  float wlv   = wl[w];
  float maxhw = ws[0];
  float lob = wlv - maxhw;
  float hib = wlv + maxhw;

  // lower_bound(lam, lob)
  int lo = 0, hi = L;
  while (lo < hi) { int m = (lo + hi) >> 1; if (lamS[m] < lob) lo = m + 1; else hi = m; }
  int start = lo;
  // upper_bound(lam, hib)
  lo = start; hi = L;
  while (lo < hi) { int m = (lo + hi) >> 1; if (lamS[m] <= hib) lo = m + 1; else hi = m; }
  int stop = lo;

  float vsum = 0.0f;
  for (int l = start; l < stop; ++l) {
    float x  = (wlv - lamS[l]) * invS[l];
    float x2 = x * x;
    if (x2 < XCUT2) {
      float exx = expf(-x2);
      float y   = yS[l];
      float y2  = y2S[l];
      float c2  = cosf(2.0f * x * y);
      float s23 = 0.0f;
#pragma unroll
      for (int n = 1; n <= 27; ++n) {
        float an = 0.5f * (float)n;
        float u  = an + x;
        float v  = an - x;
        s23 += (expf(-u * u) + expf(-v * v)) / (an * an + y2);
      }
      vsum += exx * (c2 * AS[l] + (1.0f - c2) * BS[l]) + CS[l] * s23;
    }
  }

  // blackbody * modulation epilogue
  float t  = 4700.0f * expf(ln_teff[0]);
  float q  = wlv * 1e-4f;
  float q2 = q * q;
  float q5 = q2 * q2 * q;
  float bb = 20.0f / (q5 * (expf(0.014387752f / (wlv * 1e-10f * t)) - 1.0f));
  float wn = (wlv - 10500.0f) / 2500.0f;
  float md = a_c[0] + b_c[0] * wn + c_c[0] * wn * wn;
  out[w] = (1.0f - vsum) * bb * md;
}

// ---------------------------------------------------------------------------
extern "C" void kernel_launch(void* const* d_in, const int* in_sizes, int n_in,
                              void* d_out, int out_size, void* d_ws, size_t ws_size,
                              hipStream_t stream) {
  const float* wl      = (const float*)d_in[0];
  const float* lam     = (const float*)d_in[1];
  const float* log_amp = (const float*)d_in[2];
  const float* log_sig = (const float*)d_in[3];
  const float* log_gam = (const float*)d_in[4];
  const float* ln_teff = (const float*)d_in[5];
  const float* a_c     = (const float*)d_in[6];
  const float* b_c     = (const float*)d_in[7];
  const float* c_c     = (const float*)d_in[8];
  int W = in_sizes[0];
  int L = in_sizes[1];
  float* ws = (float*)d_ws;

  phoenix_prep<<<1, 256, 0, stream>>>(lam, log_amp, log_sig, log_gam, ws, L);

  const int block = 128;
  const int grid  = (W + block - 1) / block;
  size_t shbytes  = (size_t)(7 * L) * sizeof(float) + 16;
  phoenix_main<<<grid, block, shbytes, stream>>>(wl, ws, ln_teff, a_c, b_c, c_c,
                                                 (float*)d_out, W, L);
}